// DCNv2_50371376447822
// MI455X (gfx1250) — compile-verified
//
#include <hip/hip_runtime.h>

typedef __attribute__((ext_vector_type(2))) float v2f;
typedef __attribute__((ext_vector_type(8))) float v8f;

#define HH 160
#define WW 160
#define NG 3          // groups (= kH)
#define CG 3          // channels per group
#define CT 9          // total channels in input
#define KRED 27       // 9 taps * 3 channels, padded to 28 in 7 K=4 WMMA steps
#define NJT 10        // 160 / 16 j-tiles
#define PLANE (HH * WW)
#define NE 14         // sample slots per lane: (s,v), s=0..6, v=0..1

// Compile-time helpers over the reduction index K = k*3 + c (k = tap, c = chan).
__host__ __device__ constexpr int   kc27(int K) { return K < KRED ? K : KRED - 1; }
__host__ __device__ constexpr float pyK(int K)  { return (float)(kc27(K) / 9 - 1); }          // p_y[k]
__host__ __device__ constexpr float pxK(int K)  { return (float)((kc27(K) / 3) % 3 - 1); }    // p_x[k]
__host__ __device__ constexpr int   cplK(int K) { return (kc27(K) % 3) * PLANE; }             // c * plane
__host__ __device__ constexpr int   wIdxK(int K){ return (kc27(K) / 3) * 144 + kc27(K) % 3; } // k*144 + c

// One wave per (b, g, row i). Wave computes out[b, g*16..g*16+15, i, 0..159]
// as ten 16x16 tiles, each via 7 chained V_WMMA_F32_16X16X4_F32 steps
// (full fp32: D(16o x 16j) = sum_s A_s(16o x 4) * B_s(4 x 16j)).
// Step s, slot v: K = 4s + v + 2*half ; M/N = lane&15. 27 real K + 1 pad.
__global__ __launch_bounds__(256) void dcn_wmma_kernel(
    const float* __restrict__ inp,   // (8,1,9,160,160)
    const float* __restrict__ wgt,   // (3,3,48,3)
    const float* __restrict__ off,   // (16,3,2)
    float* __restrict__ out)         // (16,48,160,160)
{
  const int lane = threadIdx.x & 31;
  const int wv = blockIdx.x * (blockDim.x >> 5) + (threadIdx.x >> 5);
  if (wv >= 16 * NG * HH) return;          // exact grid; wave-uniform guard
  const int i = wv % HH;
  const int g = (wv / HH) % NG;
  const int b = wv / (HH * NG);
  const bool half = (lane >> 4) != 0;      // lanes 16-31
  const int l15  = lane & 15;

  // Per-(b,g) fractional dilation scales (uniform over space).
  const float offy = off[(b * NG + g) * 2 + 0];
  const float offx = off[(b * NG + g) * 2 + 1];
  const float sy = 1.0f + 3.0f / offy;
  const float sx = 1.0f + 3.0f / offx;

  // ---- A matrices (weights), fp32, one 16x4 pair per step ----
  v2f Ap[7];
  const int wbase = (g * 16 + l15) * 3;    // (g*16 + M)*3
#pragma unroll
  for (int e2 = 0; e2 < NE; ++e2) {
    const int s = e2 >> 1, v = e2 & 1;
    const int K0 = 4 * s + v;              // K for half==0 ; half==1 -> +2
    int widx = wbase + (half ? wIdxK(K0 + 2) : wIdxK(K0));
    bool valid = !(half && (K0 + 2 >= KRED));
    float wval = wgt[widx];
    Ap[s][v] = valid ? wval : 0.0f;
  }

  // ---- Hoisted bilinear state per sample slot ----
  const float* p0[NE];
  const float* p1[NE];
  int   xb[NE], r0i[NE], r1i[NE];
  float w00a[NE], w01a[NE], w10a[NE], w11a[NE];
  const int pg = ((b >> 1) * CT + g * CG) * PLANE;  // input batch = b/2
#pragma unroll
  for (int e2 = 0; e2 < NE; ++e2) {
    const int s = e2 >> 1, v = e2 & 1;
    const int K0 = 4 * s + v;
    // All K-derived values are 2-way selects of folded constants.
    const float py = half ? pyK(K0 + 2) : pyK(K0);
    const float px = half ? pxK(K0 + 2) : pxK(K0);
    const int  cpl = half ? cplK(K0 + 2) : cplK(K0);
    const float kv = (half && (K0 + 2 >= KRED)) ? 0.0f : 1.0f;
    float dy = py * sy;
    float dx = px * sx;
    float yy = (float)i + dy;
    float y0f = floorf(yy);
    float wy = yy - y0f;
    int y0 = (int)y0f;
    float xx = (float)l15 + dx;            // x position at j0 = 0
    float x0f = floorf(xx);
    float wx = xx - x0f;                   // invariant as j0 steps by 16
    int x0i = min(max((int)x0f, -4), 19);  // mathematically already in [-4,19]
    xb[e2] = x0i;
    float vy0 = (y0 >= 0 && y0 < HH) ? kv : 0.0f;
    float vy1 = (y0 >= -1 && y0 < HH - 1) ? kv : 0.0f;
    int cy0 = min(max(y0, 0), HH - 1);
    int cy1 = min(max(y0 + 1, 0), HH - 1);
    int pl = pg + cpl;
    r0i[e2] = pl + cy0 * WW;               // row base (>= 0), for boundary tiles
    r1i[e2] = pl + cy1 * WW;
    p0[e2] = inp + (r0i[e2] + x0i + 16);   // interior index >= 12
    p1[e2] = inp + (r1i[e2] + x0i + 16);
    float w0 = (1.0f - wy) * vy0;
    float w1 = wy * vy1;
    w00a[e2] = w0 * (1.0f - wx);
    w01a[e2] = w0 * wx;
    w10a[e2] = w1 * (1.0f - wx);
    w11a[e2] = w1 * wx;
  }

  // out[b, g*16 + (r + 8*half), i, j0 + l15]
  float* const obp =
      out + ((b * 48 + g * 16) * HH + i) * WW + (half ? 8 * PLANE : 0) + l15;

  // 7 chained fp32 WMMA steps + store; j0 is compile-time constant at every
  // call site so (j0 + r*PLANE)*4 <= 717376 folds into the 24-bit IOFFSET.
  auto emit = [&](const float (&smp)[NE], int j0) {
    v8f acc = {};
#pragma unroll
    for (int s = 0; s < 7; ++s) {
      v2f bb;
      bb[0] = smp[2 * s];
      bb[1] = smp[2 * s + 1];
      acc = __builtin_amdgcn_wmma_f32_16x16x4_f32(
          false, Ap[s], false, bb, (short)0, acc, false, false);
    }
#pragma unroll
    for (int r = 0; r < 8; ++r)
      obp[j0 + r * PLANE] = acc[r];
  };

  // One-sided boundary tile: left (j0=0, x0 in [-4,19], only low clamp) or
  // right (j0=144, x0 in [140,163], only high clamp).
  auto boundary = [&](int j0, bool left) {
    float smp[NE];
#pragma unroll
    for (int e2 = 0; e2 < NE; ++e2) {
      int x0 = xb[e2] + j0;
      float vx0, vx1;
      int cx0, cx1;
      if (left) {
        vx0 = (x0 >= 0)  ? 1.0f : 0.0f;
        vx1 = (x0 >= -1) ? 1.0f : 0.0f;
        cx0 = max(x0, 0);
        cx1 = max(x0 + 1, 0);
      } else {
        vx0 = (x0 < WW)     ? 1.0f : 0.0f;
        vx1 = (x0 + 1 < WW) ? 1.0f : 0.0f;
        cx0 = min(x0, WW - 1);
        cx1 = min(x0 + 1, WW - 1);
      }
      float f00 = inp[r0i[e2] + cx0];
      float f01 = inp[r0i[e2] + cx1];
      float f10 = inp[r1i[e2] + cx0];
      float f11 = inp[r1i[e2] + cx1];
      smp[e2] = (w00a[e2] * vx0) * f00 + (w01a[e2] * vx1) * f01
              + (w10a[e2] * vx0) * f10 + (w11a[e2] * vx1) * f11;
    }
    emit(smp, j0);
  };

  boundary(0, true);                       // left boundary tile

#pragma unroll
  for (int t = 0; t < NJT - 2; ++t) {      // 8 interior tiles, j0 = 16..128
    const int joff = t * 16;               // constant after unroll
    float smp[NE];
#pragma unroll
    for (int e2 = 0; e2 < NE; ++e2) {
      // loads at hoisted base + constant immediate offset (b64-merged pairs)
      smp[e2] = w00a[e2] * p0[e2][joff] + w01a[e2] * p0[e2][joff + 1]
              + w10a[e2] * p1[e2][joff] + w11a[e2] * p1[e2][joff + 1];
    }
    emit(smp, joff + 16);
  }

  boundary((NJT - 1) * 16, false);         // right boundary tile
}

extern "C" void kernel_launch(void* const* d_in, const int* in_sizes, int n_in,
                              void* d_out, int out_size, void* d_ws, size_t ws_size,
                              hipStream_t stream) {
  const float* inp = (const float*)d_in[0];
  const float* wgt = (const float*)d_in[1];
  const float* off = (const float*)d_in[2];
  float* out = (float*)d_out;

  const int waves = 16 * NG * HH;           // 7680 waves, one per (b,g,i)
  const int wavesPerBlock = 8;              // 256 threads (wave32)
  const int blocks = waves / wavesPerBlock; // 960
  dcn_wmma_kernel<<<blocks, 256, 0, stream>>>(inp, wgt, off, out);
}